// GraphConvolution_11871289606689
// MI455X (gfx1250) — compile-verified
//
#include <hip/hip_runtime.h>
#include <math.h>

#define N_NODES 50000
#define N_EDGES 800000
#define DIM     128

typedef __attribute__((ext_vector_type(2))) float v2f;
typedef __attribute__((ext_vector_type(8))) float v8f;

// ---------------------------------------------------------------------------
// Kernel 1: zero the hi accumulator (d_ws is poisoned by the harness).
// ---------------------------------------------------------------------------
__global__ void gcn_zero_kernel(float4* __restrict__ ws, int n4) {
    int t = blockIdx.x * blockDim.x + threadIdx.x;
    if (t < n4) ws[t] = make_float4(0.f, 0.f, 0.f, 0.f);
}

// ---------------------------------------------------------------------------
// Kernel 2: COO SpMM scatter: hi[row[e], :] += vals[e] * input[col[e], :]
// 32 lanes per edge, each lane handles 4 consecutive features via float4.
// input + hi are L2-resident (25.6 MB each, 192 MB L2); atomics resolve at L2.
// ---------------------------------------------------------------------------
__global__ void __launch_bounds__(256)
gcn_spmm_scatter_kernel(const float* __restrict__ input,
                        const float* __restrict__ vals,
                        const int*   __restrict__ row,
                        const int*   __restrict__ col,
                        float*       __restrict__ hi) {
    long t = (long)blockIdx.x * blockDim.x + threadIdx.x;   // E*32 threads
    int e = (int)(t >> 5);
    int q = (int)(t & 31);                                  // feature quad id
    if (e >= N_EDGES) return;

    float v = vals[e];
    int   c = col[e];
    int   r = row[e];

    const float4 x = *(const float4*)(input + (size_t)c * DIM + 4 * q);
    float* dst = hi + (size_t)r * DIM + 4 * q;

    __hip_atomic_fetch_add(dst + 0, v * x.x, __ATOMIC_RELAXED, __HIP_MEMORY_SCOPE_AGENT);
    __hip_atomic_fetch_add(dst + 1, v * x.y, __ATOMIC_RELAXED, __HIP_MEMORY_SCOPE_AGENT);
    __hip_atomic_fetch_add(dst + 2, v * x.z, __ATOMIC_RELAXED, __HIP_MEMORY_SCOPE_AGENT);
    __hip_atomic_fetch_add(dst + 3, v * x.w, __ATOMIC_RELAXED, __HIP_MEMORY_SCOPE_AGENT);
}

// ---------------------------------------------------------------------------
// Kernel 3: fused support + GEMM + residual epilogue.
//   support = (1-alpha)*hi + alpha*h0              (staged into LDS, 16x128)
//   out     = theta * support @ W + (1-theta)*support
// One block = 16 node rows; 8 waves, wave w computes the 16x16 output tile
// at columns [16w,16w+16) via 32x V_WMMA_F32_16X16X4_F32 over K=128.
//
// 16x4 f32 A layout (ISA 7.12.2): lanes 0-15 -> M=lane, VGPR0=K0,VGPR1=K1;
// lanes 16-31 -> M=lane-16, VGPR0=K2,VGPR1=K3.  B is the mirrored layout
// with N on lanes and K split across (VGPR index, lane half).
// ---------------------------------------------------------------------------
__global__ void __launch_bounds__(256)
gcn_gemm_epilogue_kernel(const float* __restrict__ hi,
                         const float* __restrict__ h0,
                         const float* __restrict__ W,
                         const float* __restrict__ lamda_p,
                         const float* __restrict__ alpha_p,
                         const int*   __restrict__ l_p,
                         float*       __restrict__ out) {
    __shared__ float S[16][132];   // pad 128->132: conflict-free strided reads

    const float alpha = alpha_p[0];
    const float theta = logf(lamda_p[0] / (float)l_p[0] + 1.0f);
    const float oma   = 1.0f - alpha;
    const float omt   = 1.0f - theta;

    const int base_row = blockIdx.x * 16;    // 50000 = 3125 * 16 exactly

    // Stage support tile: 16x128 f32 = 512 float4, 2 per thread.
    for (int j = threadIdx.x; j < 512; j += 256) {
        int r = j >> 5;
        int q = j & 31;
        size_t g = (size_t)(base_row + r) * DIM + 4 * q;
        const float4 a = *(const float4*)(hi + g);
        const float4 b = *(const float4*)(h0 + g);
        float4 s;
        s.x = oma * a.x + alpha * b.x;
        s.y = oma * a.y + alpha * b.y;
        s.z = oma * a.z + alpha * b.z;
        s.w = oma * a.w + alpha * b.w;
        *(float4*)&S[r][4 * q] = s;
    }
    __syncthreads();

    const int lane = threadIdx.x & 31;
    const int wave = threadIdx.x >> 5;   // 0..7 -> output column tile
    const int half = lane >> 4;          // lane half selects K pair
    const int lid  = lane & 15;
    const int n0   = wave * 16;

    v8f c = {0.f, 0.f, 0.f, 0.f, 0.f, 0.f, 0.f, 0.f};

    #pragma unroll 4
    for (int ks = 0; ks < 32; ++ks) {
        const int k0 = ks * 4 + 2 * half;
        v2f a, b;
        a[0] = S[lid][k0];
        a[1] = S[lid][k0 + 1];
        b[0] = W[(size_t)k0 * DIM + n0 + lid];
        b[1] = W[(size_t)(k0 + 1) * DIM + n0 + lid];
        c = __builtin_amdgcn_wmma_f32_16x16x4_f32(
                /*neg_a=*/false, a, /*neg_b=*/false, b,
                /*c_mod=*/(short)0, c, /*reuse_a=*/false, /*reuse_b=*/false);
    }

    // Epilogue: D layout (16x16 f32): lanes 0-15 -> M = v, lanes 16-31 -> M = v+8.
    #pragma unroll
    for (int v = 0; v < 8; ++v) {
        int m = v + 8 * half;
        float sv = S[m][n0 + lid];
        out[(size_t)(base_row + m) * DIM + n0 + lid] = theta * c[v] + omt * sv;
    }
}

// ---------------------------------------------------------------------------
extern "C" void kernel_launch(void* const* d_in, const int* in_sizes, int n_in,
                              void* d_out, int out_size, void* d_ws, size_t ws_size,
                              hipStream_t stream) {
    (void)in_sizes; (void)n_in; (void)out_size; (void)ws_size;

    const float* input  = (const float*)d_in[0];   // [N, D]
    const float* h0     = (const float*)d_in[1];   // [N, D]
    const float* vals   = (const float*)d_in[2];   // [E]
    const float* weight = (const float*)d_in[3];   // [D, D]
    const float* lamda  = (const float*)d_in[4];   // scalar
    const float* alpha  = (const float*)d_in[5];   // scalar
    const int*   row    = (const int*)d_in[6];     // [E]
    const int*   col    = (const int*)d_in[7];     // [E]
    const int*   l      = (const int*)d_in[8];     // scalar int

    float* out = (float*)d_out;                    // [N, D]
    float* hi  = (float*)d_ws;                     // [N, D] accumulator

    // 1) zero hi
    int n4 = N_NODES * DIM / 4;
    gcn_zero_kernel<<<(n4 + 255) / 256, 256, 0, stream>>>((float4*)hi, n4);

    // 2) edge scatter (E * 32 lanes)
    long total = (long)N_EDGES * 32;
    gcn_spmm_scatter_kernel<<<(int)((total + 255) / 256), 256, 0, stream>>>(
        input, vals, row, col, hi);

    // 3) fused support + WMMA GEMM + residual
    gcn_gemm_epilogue_kernel<<<N_NODES / 16, 256, 0, stream>>>(
        hi, h0, weight, lamda, alpha, l, out);
}